// obj_decoder_15564961481391
// MI455X (gfx1250) — compile-verified
//
#include <hip/hip_runtime.h>
#include <hip/hip_bf16.h>
#include <math.h>

typedef float v2f __attribute__((ext_vector_type(2)));
typedef float v8f __attribute__((ext_vector_type(8)));

// ---- problem shape (fixed by the reference) ----
#define BB 32
#define CC 64
#define NN 2048
#define DD 6

static constexpr float LOG_2PI_F = 1.8378770664093453f;   // log(2*pi)
static constexpr float DUMMY_F   = -4.605170185988091f;   // -2*log(10)
static constexpr float EPS_F     = 1e-16f;

// ---- flat output offsets (reference tuple order, all-as-float) ----
static constexpr size_t S_BCN  = (size_t)BB * CC * NN;        // 4,194,304
static constexpr size_t S_BN   = (size_t)BB * NN;             // 65,536
static constexpr size_t S_BN6  = S_BN * DD;                   // 393,216
static constexpr size_t S_BC1N = (size_t)BB * (CC + 1) * NN;  // 4,259,840

static constexpr size_t O_LOGPROB = 0;                  // scalar
static constexpr size_t O_VPRES   = O_LOGPROB + 1;      // [B,C,N] bool->float
static constexpr size_t O_WIN     = O_VPRES + S_BCN;    // [B,N,6]
static constexpr size_t O_WINP    = O_WIN + S_BN6;      // [B,N]
static constexpr size_t O_ISFROM  = O_WINP + S_BN;      // [B,N] int->float
static constexpr size_t O_MLOG    = O_ISFROM + S_BN;    // [B,C+1,N]
static constexpr size_t O_MLP     = O_MLOG + S_BC1N;    // [B,C+1,N]
static constexpr size_t O_SOFTW   = O_MLP + S_BC1N;     // [B,N,6]
static constexpr size_t O_SOFTWP  = O_SOFTW + S_BN6;    // [B,N]
static constexpr size_t O_PMP     = O_SOFTWP + S_BN;    // [B,N,C]
static constexpr size_t O_CAPS    = O_PMP + S_BCN;      // [B,C]

#define TPB 128   // threads per block, one point per thread

// Wave32 full-lane sum using the matrix pipe for the +16 butterfly
// (V_WMMA_F32_16X16X4_F32 with A = ones sums each B column: lanes l and l+16),
// then shfl_xor finishes the 16-wide tree. f32 end to end.
__device__ __forceinline__ float wave_sum_wmma(float v) {
    v2f a_ones = {1.0f, 1.0f};
    v2f b_val  = {v, 0.0f};
    v8f c_zero = {0.f, 0.f, 0.f, 0.f, 0.f, 0.f, 0.f, 0.f};
    v8f d = __builtin_amdgcn_wmma_f32_16x16x4_f32(
        /*neg_a=*/false, a_ones, /*neg_b=*/false, b_val,
        /*c_mod=*/(short)0, c_zero, /*reuse_a=*/false, /*reuse_b=*/false);
    float t = d[0];                 // lane l holds v[l&15] + v[(l&15)+16]
    t += __shfl_xor(t, 8, 32);
    t += __shfl_xor(t, 4, 32);
    t += __shfl_xor(t, 2, 32);
    t += __shfl_xor(t, 1, 32);
    return t;                       // full 32-lane sum in every lane
}

__global__ __launch_bounds__(TPB)
void capsule_mixture_kernel(const float* __restrict__ x,
                            const float* __restrict__ votes,
                            const float* __restrict__ scale,
                            const float* __restrict__ presp,
                            const float* __restrict__ dvote,
                            float* __restrict__ out,
                            float* __restrict__ partial)
{
    extern __shared__ float smem[];
    float* post_s = smem;                 // TPB * 65 floats, stride 65 (odd -> no bank conflicts)
    float* lse_s  = smem + TPB * (CC + 1);
    __shared__ float wred[TPB / 32];

    const int tid = threadIdx.x;
    const int p   = blockIdx.x * TPB + tid;   // global point id, 0..B*N-1
    const int b   = p >> 11;                  // N = 2048
    const int n   = p & (NN - 1);
    const size_t bn = (size_t)b * NN + n;
    const size_t cb = (size_t)b * CC * NN + n;   // + c*NN -> [b,c,n]

    // x[b,n,0:6]   (24B rows -> 8B aligned, three b64 loads)
    const float2* xv = reinterpret_cast<const float2*>(x + bn * DD);
    const float2 x01 = xv[0], x23 = xv[1], x45 = xv[2];

    // ---------- pass 1: stream votes from HBM ----------
    float maxpo = -3.0e38f;
    int   amax  = 0;
    float wv0 = 0.f, wv1 = 0.f, wv2 = 0.f, wv3 = 0.f, wv4 = 0.f, wv5 = 0.f;
    float wpr = 0.f;
    // online logsumexp of mixing logits, seeded with the dummy logit
    float mml = DUMMY_F;
    float sml = 1.0f;

    #pragma unroll 4
    for (int c = 0; c < CC; ++c) {
        const size_t idx = cb + (size_t)c * NN;
        const float2* vv = reinterpret_cast<const float2*>(votes + idx * DD);
        const float2 a01 = vv[0], a23 = vv[1], a45 = vv[2];
        if (c + 1 < CC)   // gfx1250 global_prefetch_b8: next capsule's vote row
            __builtin_prefetch(votes + (idx + NN) * DD, 0, 0);
        const float s  = scale[idx];
        const float pr = presp[idx];
        const float inv = 1.0f / s;
        float z, acc = 0.f;
        z = (x01.x - a01.x) * inv; acc = fmaf(z, z, acc);
        z = (x01.y - a01.y) * inv; acc = fmaf(z, z, acc);
        z = (x23.x - a23.x) * inv; acc = fmaf(z, z, acc);
        z = (x23.y - a23.y) * inv; acc = fmaf(z, z, acc);
        z = (x45.x - a45.x) * inv; acc = fmaf(z, z, acc);
        z = (x45.y - a45.y) * inv; acc = fmaf(z, z, acc);
        const float vlp = fmaf(-0.5f, acc, fmaf(-(float)DD, logf(s), -0.5f * DD * LOG_2PI_F));
        const float ml  = logf(pr + EPS_F);
        out[O_MLOG + ((size_t)(b * (CC + 1) + c)) * NN + n] = ml;         // mixing_logits
        out[O_VPRES + idx] = (DUMMY_F < ml) ? 1.0f : 0.0f;                // vote_presence
        const float po = ml + vlp;
        post_s[tid * (CC + 1) + c] = po;
        if (po > maxpo) {
            maxpo = po; amax = c; wpr = pr;
            wv0 = a01.x; wv1 = a01.y; wv2 = a23.x; wv3 = a23.y; wv4 = a45.x; wv5 = a45.y;
        }
        // online LSE for mixing logits
        if (ml > mml) { sml = fmaf(sml, expf(mml - ml), 1.0f); mml = ml; }
        else          { sml += expf(ml - mml); }
    }

    // dummy mixture component
    const float po_d = DUMMY_F + DUMMY_F;           // dummy logit + dummy vote_log_prob
    post_s[tid * (CC + 1) + CC] = po_d;
    const float maxall = fmaxf(maxpo, po_d);
    const float lse_ml = mml + logf(sml);
    out[O_MLOG + ((size_t)(b * (CC + 1) + CC)) * NN + n] = DUMMY_F;
    out[O_MLP  + ((size_t)(b * (CC + 1) + CC)) * NN + n] = DUMMY_F - lse_ml;

    // posterior logsumexp over 65 entries (LDS-resident)
    float sume = 0.f;
    #pragma unroll 5
    for (int c = 0; c <= CC; ++c)
        sume += expf(post_s[tid * (CC + 1) + c] - maxall);
    const float lse_post = maxall + logf(sume);
    lse_s[tid] = lse_post;

    // winner outputs
    {
        float* w = out + O_WIN + bn * DD;
        w[0] = wv0; w[1] = wv1; w[2] = wv2; w[3] = wv3; w[4] = wv4; w[5] = wv5;
        out[O_WINP   + bn] = wpr;
        out[O_ISFROM + bn] = (float)(amax / NN);   // reference: idx // N  (== 0 here)
    }

    // soft winner accumulators, dummy term first
    const float pmpd = expf(po_d - lse_post);
    const float2* dv = reinterpret_cast<const float2*>(dvote + (size_t)n * DD);
    const float2 d01 = dv[0], d23 = dv[1], d45 = dv[2];
    float sw0 = pmpd * d01.x, sw1 = pmpd * d01.y, sw2 = pmpd * d23.x;
    float sw3 = pmpd * d23.y, sw4 = pmpd * d45.x, sw5 = pmpd * d45.y;
    float swp = 0.f;

    // ---------- pass 2: re-read votes (L2-resident: 100MB < 192MB L2) ----------
    #pragma unroll 4
    for (int c = 0; c < CC; ++c) {
        const size_t idx = cb + (size_t)c * NN;
        const float pmp = expf(post_s[tid * (CC + 1) + c] - lse_post);
        const float2* vv = reinterpret_cast<const float2*>(votes + idx * DD);
        const float2 a01 = vv[0], a23 = vv[1], a45 = vv[2];
        sw0 = fmaf(pmp, a01.x, sw0); sw1 = fmaf(pmp, a01.y, sw1);
        sw2 = fmaf(pmp, a23.x, sw2); sw3 = fmaf(pmp, a23.y, sw3);
        sw4 = fmaf(pmp, a45.x, sw4); sw5 = fmaf(pmp, a45.y, sw5);
        const float pr = presp[idx];
        swp = fmaf(pmp, pr, swp);
        out[O_MLP + ((size_t)(b * (CC + 1) + c)) * NN + n] = logf(pr + EPS_F) - lse_ml;
    }
    {
        float* w = out + O_SOFTW + bn * DD;
        w[0] = sw0; w[1] = sw1; w[2] = sw2; w[3] = sw3; w[4] = sw4; w[5] = sw5;
        out[O_SOFTWP + bn] = swp;
    }

    // ---------- cooperative, coalesced [B,N,C] softmax write ----------
    __syncthreads();
    const int n0 = (blockIdx.x * TPB) & (NN - 1);     // block covers a single b
    const size_t base9 = O_PMP + ((size_t)b * NN + n0) * CC;
    #pragma unroll 4
    for (int k = tid; k < TPB * CC; k += TPB) {
        const int nl = k >> 6;           // local point
        const int c  = k & (CC - 1);
        out[base9 + k] = expf(post_s[nl * (CC + 1) + c] - lse_s[nl]);
    }

    // ---------- log_prob partial: WMMA wave sum + cross-wave LDS ----------
    const float wsum = wave_sum_wmma(lse_post);
    if ((tid & 31) == 0) wred[tid >> 5] = wsum;
    __syncthreads();
    if (tid == 0) {
        float t = 0.f;
        #pragma unroll
        for (int w = 0; w < TPB / 32; ++w) t += wred[w];
        partial[blockIdx.x] = t;
    }
}

// caps_presence_prob[b,c] = max over n of presence
__global__ __launch_bounds__(256)
void caps_max_kernel(const float* __restrict__ presp, float* __restrict__ out)
{
    __shared__ float s[256];
    const int bc = blockIdx.x;            // 0..B*C-1
    const int t  = threadIdx.x;
    const float* p = presp + (size_t)bc * NN;
    float m = -3.0e38f;
    for (int i = t; i < NN; i += 256) m = fmaxf(m, p[i]);
    s[t] = m;
    __syncthreads();
    for (int off = 128; off > 0; off >>= 1) {
        if (t < off) s[t] = fmaxf(s[t], s[t + off]);
        __syncthreads();
    }
    if (t == 0) out[O_CAPS + bc] = s[0];
}

// sum 512 block partials, mean over batch
__global__ __launch_bounds__(512)
void finalize_kernel(const float* __restrict__ partial, float* __restrict__ out)
{
    __shared__ float s[512];
    const int t = threadIdx.x;
    s[t] = partial[t];
    __syncthreads();
    for (int off = 256; off > 0; off >>= 1) {
        if (t < off) s[t] += s[t + off];
        __syncthreads();
    }
    if (t == 0) out[O_LOGPROB] = s[0] * (1.0f / (float)BB);
}

extern "C" void kernel_launch(void* const* d_in, const int* in_sizes, int n_in,
                              void* d_out, int out_size, void* d_ws, size_t ws_size,
                              hipStream_t stream)
{
    (void)in_sizes; (void)n_in; (void)out_size; (void)ws_size;
    const float* x     = (const float*)d_in[0];
    const float* votes = (const float*)d_in[1];
    const float* scale = (const float*)d_in[2];
    const float* presp = (const float*)d_in[3];
    const float* dvote = (const float*)d_in[4];
    float* out     = (float*)d_out;
    float* partial = (float*)d_ws;           // 512 floats

    const int nblocks = (BB * NN) / TPB;     // 512
    const size_t shbytes = (size_t)(TPB * (CC + 1) + TPB) * sizeof(float);  // 33,792 B

    capsule_mixture_kernel<<<nblocks, TPB, shbytes, stream>>>(
        x, votes, scale, presp, dvote, out, partial);
    caps_max_kernel<<<BB * CC, 256, 0, stream>>>(presp, out);
    finalize_kernel<<<1, nblocks, 0, stream>>>(partial, out);
}